// DecouplingSharedSpecialLoss_7060926235201
// MI455X (gfx1250) — compile-verified
//
#include <hip/hip_runtime.h>

typedef __attribute__((ext_vector_type(2))) float v2f;
typedef __attribute__((ext_vector_type(8))) float v8f;

#define DIM 2048
#define GROUPS 1024            // 16384 rows / 16 rows-per-group
#define WAVES_PER_GROUP 4      // K split 4 ways -> 4096 waves total
#define K_PER_WAVE (DIM / WAVES_PER_GROUP)   // 512
#define GROUPS_PER_BLOCK 2     // 256 threads = 8 waves = 2 groups

// Stage 1: per group of 16 rows, compute S = A x B^T (16x16, f32 WMMA),
// per-row squared norms, then block-diagonal clamped -log(1-sim) partial sums.
__global__ __launch_bounds__(256)
void sim_loss_partial_kernel(const float* __restrict__ A,
                             const float* __restrict__ B,
                             float* __restrict__ ws)
{
    const int lane  = threadIdx.x & 31;
    const int wib   = threadIdx.x >> 5;        // wave in block: 0..7
    const int slot  = wib >> 2;                // group slot in block: 0..1
    const int winG  = wib & 3;                 // wave in group: 0..3 (K quarter)
    const int group = blockIdx.x * GROUPS_PER_BLOCK + slot;

    const int rowInGroup = lane & 15;
    const int colOff     = (lane >> 4) << 1;   // 0 or 2 (WMMA f32 A/B layout)
    const size_t rowBase = (size_t)(group * 16 + rowInGroup) * DIM;

    const float* pa = A + rowBase + (size_t)winG * K_PER_WAVE + colOff;
    const float* pb = B + rowBase + (size_t)winG * K_PER_WAVE + colOff;

    v8f c = {};                 // 16x16 f32 accumulator (S_ab partial)
    float sqa = 0.0f, sqb = 0.0f;

    #pragma unroll 8
    for (int k = 0; k < K_PER_WAVE; k += 4) {
        v2f va = *(const v2f*)(pa + k);
        v2f vb = *(const v2f*)(pb + k);
        sqa = fmaf(va.x, va.x, sqa);
        sqa = fmaf(va.y, va.y, sqa);
        sqb = fmaf(vb.x, vb.x, sqb);
        sqb = fmaf(vb.y, vb.y, sqb);
        // D = A(16x4) x B(4x16) + C ; lane l holds row (l&15), K = k + colOff + {0,1}
        c = __builtin_amdgcn_wmma_f32_16x16x4_f32(
                /*neg_a=*/false, va, /*neg_b=*/false, vb,
                /*c_mod=*/(short)0, c, /*reuse_a=*/false, /*reuse_b=*/false);
    }

    // Combine the 4 K-quarter waves of this group through LDS (deterministic order).
    __shared__ float comb[GROUPS_PER_BLOCK][WAVES_PER_GROUP][32][10];
    #pragma unroll
    for (int r = 0; r < 8; ++r) comb[slot][winG][lane][r] = c[r];
    comb[slot][winG][lane][8] = sqa;
    comb[slot][winG][lane][9] = sqb;
    __syncthreads();

    if (winG == 0) {
        float cs[8];
        #pragma unroll
        for (int r = 0; r < 8; ++r) cs[r] = 0.0f;
        float tsqa = 0.0f, tsqb = 0.0f;
        #pragma unroll
        for (int w = 0; w < WAVES_PER_GROUP; ++w) {
            #pragma unroll
            for (int r = 0; r < 8; ++r) cs[r] += comb[slot][w][lane][r];
            tsqa += comb[slot][w][lane][8];
            tsqb += comb[slot][w][lane][9];
        }

        // lane l and lane l^16 jointly cover row (l&15); sum the halves.
        tsqa += __shfl_xor(tsqa, 16);
        tsqb += __shfl_xor(tsqb, 16);
        const float na = fmaxf(sqrtf(tsqa), 1e-12f);  // ||a_{lane&15}||
        const float nb = fmaxf(sqrtf(tsqb), 1e-12f);  // ||b_{lane&15}||

        // C/D layout: VGPR r, lanes 0-15 -> (M=r, N=lane); lanes 16-31 -> (M=r+8, N=lane-16)
        const int N = lane & 15;
        float acc = 0.0f;
        #pragma unroll
        for (int r = 0; r < 8; ++r) {
            const float nA_lo = __shfl(na, r);
            const float nA_hi = __shfl(na, r + 8);
            const float nA    = (lane < 16) ? nA_lo : nA_hi;
            const int   M     = (lane < 16) ? r : (r + 8);
            if ((M >> 2) == (N >> 2)) {            // block-diagonal 4x4 tiles only
                float sim = cs[r] / (nA * nb);
                sim = fminf(fmaxf(sim, 0.0005f), 0.9995f);
                acc += -log1pf(-sim);              // -log(1 - sim)
            }
        }
        acc *= (1.0f / 16.0f);                     // mean over 16 entries per chunk

        #pragma unroll
        for (int off = 16; off > 0; off >>= 1)
            acc += __shfl_xor(acc, off);
        if (lane == 0) ws[group] = acc;
    }
}

// Stage 2: deterministic single-block tree reduction of the 1024 group partials.
__global__ __launch_bounds__(256)
void reduce_partials_kernel(const float* __restrict__ ws, float* __restrict__ out)
{
    __shared__ float s[256];
    const int t = threadIdx.x;
    float v = ws[t] + ws[t + 256] + ws[t + 512] + ws[t + 768];
    s[t] = v;
    __syncthreads();
    #pragma unroll
    for (int stride = 128; stride > 0; stride >>= 1) {
        if (t < stride) s[t] += s[t + stride];
        __syncthreads();
    }
    if (t == 0) out[0] = s[0];
}

extern "C" void kernel_launch(void* const* d_in, const int* in_sizes, int n_in,
                              void* d_out, int out_size, void* d_ws, size_t ws_size,
                              hipStream_t stream)
{
    const float* shared_features   = (const float*)d_in[0];
    const float* specific_features = (const float*)d_in[1];
    float* out = (float*)d_out;
    float* ws  = (float*)d_ws;   // needs GROUPS * 4 bytes = 4 KB

    const int blocks = GROUPS / GROUPS_PER_BLOCK;   // 512
    sim_loss_partial_kernel<<<blocks, 256, 0, stream>>>(shared_features, specific_features, ws);
    reduce_partials_kernel<<<1, 256, 0, stream>>>(ws, out);
}